// GATClassifier_30313879175196
// MI455X (gfx1250) — compile-verified
//
#include <hip/hip_runtime.h>

#define N_NODES    100000
#define N_EDGES    3200000
#define D_IN       200
#define HEADS      5
#define F_OUT      10
#define HF         50        // HEADS * F_OUT
#define HF_PAD     64
#define H_PAD      8
#define NUM_GRAPHS 1000
#define NEG_SLOPE  0.2f
#define E_TOT      (N_EDGES + N_NODES)   // edges + self loops

typedef __attribute__((ext_vector_type(2))) float v2f;
typedef __attribute__((ext_vector_type(8))) float v8f;

// ---- float <-> order-preserving uint (for atomic segment-max) ----
__device__ __forceinline__ unsigned f_ordered(float f) {
    unsigned u = __float_as_uint(f);
    return (u & 0x80000000u) ? ~u : (u | 0x80000000u);
}
__device__ __forceinline__ float f_unordered(unsigned u) {
    return __uint_as_float((u & 0x80000000u) ? (u & 0x7FFFFFFFu) : ~u);
}

// =====================================================================
// Kernel 1: xw[n, c] = sum_d x[n,d] * W[h, d, f]   (c = h*F_OUT + f)
// One wave per 16x16 tile, fp32 WMMA, K stepped by 4 (V_WMMA_F32_16X16X4_F32).
// Block = 128 threads = 4 waves, each wave owns one of the 4 n-tiles (64 padded
// cols). Invalid columns (>=50) are handled branch-free: clamped B index + 0-mask
// multiply, so the inner loop has no EXEC manipulation around the WMMA.
// =====================================================================
__global__ __launch_bounds__(128) void gat_gemm_xw(const float* __restrict__ x,
                                                   const float* __restrict__ W,
                                                   float* __restrict__ xw) {
    const int lane = threadIdx.x & 31;
    const int wave = threadIdx.x >> 5;          // n-tile 0..3
    const int m0   = blockIdx.x * 16;
    const int n0   = wave * 16;
    const int half = lane >> 4;                 // 0 => K{0,1}, 1 => K{2,3}
    const int l15  = lane & 15;

    const int row  = m0 + l15;                  // N_NODES % 16 == 0, no mask needed
    const int col  = n0 + l15;
    const bool colOK = (col < HF);
    const float bmask = colOK ? 1.0f : 0.0f;    // zero out padded columns
    int h = colOK ? (col / F_OUT) : 0;
    int f = colOK ? (col - h * F_OUT) : 0;
    const int wbase = h * (D_IN * F_OUT) + f;   // W[h, k, f] = W[wbase + k*F_OUT]

    const float* xrow = x + (long)row * D_IN + half * 2;

    v8f acc = {};
    #pragma unroll 2
    for (int k0 = 0; k0 < D_IN; k0 += 4) {
        __builtin_prefetch(xrow + k0 + 16, 0, 1);       // speculative; OOB dropped
        float2 av = *(const float2*)(xrow + k0);        // 8B-aligned (even offsets)
        v2f a; a.x = av.x; a.y = av.y;
        int kb = k0 + half * 2;
        v2f b;
        b.x = W[wbase + kb * F_OUT] * bmask;
        b.y = W[wbase + (kb + 1) * F_OUT] * bmask;
        acc = __builtin_amdgcn_wmma_f32_16x16x4_f32(false, a, false, b,
                                                    (short)0, acc, false, false);
    }
    // C/D layout: VGPR i, lanes 0-15 -> M=i, lanes 16-31 -> M=i+8; N = lane&15
    float* outp = xw + (long)(m0 + half * 8) * HF_PAD + (n0 + l15);
    #pragma unroll
    for (int i = 0; i < 8; ++i) outp[(long)i * HF_PAD] = acc[i];
}

// =====================================================================
// Kernel 2: a_s[n,h] = <xw[n,h,:], att_src[h,:]> ; same for a_d
// =====================================================================
__global__ void gat_attn(const float* __restrict__ xw,
                         const float* __restrict__ att_src,
                         const float* __restrict__ att_dst,
                         float* __restrict__ a_s, float* __restrict__ a_d) {
    int n = blockIdx.x * blockDim.x + threadIdx.x;
    if (n >= N_NODES) return;
    const float* r = xw + (long)n * HF_PAD;
    #pragma unroll
    for (int h = 0; h < HEADS; ++h) {
        float s = 0.0f, d = 0.0f;
        #pragma unroll
        for (int f = 0; f < F_OUT; ++f) {
            float v = r[h * F_OUT + f];
            s += v * att_src[h * F_OUT + f];
            d += v * att_dst[h * F_OUT + f];
        }
        a_s[(long)n * H_PAD + h] = s;
        a_d[(long)n * H_PAD + h] = d;
    }
}

// =====================================================================
// Zero-fill (init for atomic targets; ordered-uint 0 == -inf)
// =====================================================================
__global__ void gat_zero(float* __restrict__ p, long n) {
    long i = blockIdx.x * (long)blockDim.x + threadIdx.x;
    long stride = (long)gridDim.x * blockDim.x;
    for (; i < n; i += stride) p[i] = 0.0f;
}

__device__ __forceinline__ void edge_nodes(const int* __restrict__ ei, long e,
                                           int& s, int& d) {
    if (e < N_EDGES) { s = ei[e]; d = ei[N_EDGES + e]; }
    else             { s = d = (int)(e - N_EDGES); }          // self loop
}

// =====================================================================
// Kernel 3: segment-max of leaky_relu scores (atomic max, ordered-uint)
// =====================================================================
__global__ void gat_edge_max(const int* __restrict__ ei,
                             const float* __restrict__ a_s,
                             const float* __restrict__ a_d,
                             unsigned* __restrict__ m_u) {
    long e = blockIdx.x * (long)blockDim.x + threadIdx.x;
    if (e >= E_TOT) return;
    int s, d; edge_nodes(ei, e, s, d);
    #pragma unroll
    for (int h = 0; h < HEADS; ++h) {
        float sc = a_s[(long)s * H_PAD + h] + a_d[(long)d * H_PAD + h];
        sc = sc > 0.0f ? sc : NEG_SLOPE * sc;
        atomicMax(&m_u[(long)d * H_PAD + h], f_ordered(sc));
    }
}

// =====================================================================
// Kernel 4: denom[d,h] += exp(score - m[d,h])
// =====================================================================
__global__ void gat_edge_denom(const int* __restrict__ ei,
                               const float* __restrict__ a_s,
                               const float* __restrict__ a_d,
                               const unsigned* __restrict__ m_u,
                               float* __restrict__ denom) {
    long e = blockIdx.x * (long)blockDim.x + threadIdx.x;
    if (e >= E_TOT) return;
    int s, d; edge_nodes(ei, e, s, d);
    #pragma unroll
    for (int h = 0; h < HEADS; ++h) {
        float sc = a_s[(long)s * H_PAD + h] + a_d[(long)d * H_PAD + h];
        sc = sc > 0.0f ? sc : NEG_SLOPE * sc;
        float m = f_unordered(m_u[(long)d * H_PAD + h]);
        atomicAdd(&denom[(long)d * H_PAD + h], expf(sc - m));
    }
}

// =====================================================================
// Kernel 5: agg[d, h*F_OUT+f] += alpha(e,h) * xw[s, h*F_OUT+f]
// one thread per (edge, head); 10 atomic f32 adds each
// =====================================================================
__global__ void gat_edge_agg(const int* __restrict__ ei,
                             const float* __restrict__ a_s,
                             const float* __restrict__ a_d,
                             const unsigned* __restrict__ m_u,
                             const float* __restrict__ denom,
                             const float* __restrict__ xw,
                             float* __restrict__ agg) {
    long tid = blockIdx.x * (long)blockDim.x + threadIdx.x;
    if (tid >= (long)E_TOT * HEADS) return;
    long e = tid / HEADS;
    int  h = (int)(tid - e * HEADS);
    int s, d; edge_nodes(ei, e, s, d);
    float sc = a_s[(long)s * H_PAD + h] + a_d[(long)d * H_PAD + h];
    sc = sc > 0.0f ? sc : NEG_SLOPE * sc;
    float m = f_unordered(m_u[(long)d * H_PAD + h]);
    float alpha = expf(sc - m) / denom[(long)d * H_PAD + h];
    const float* xs = xw + (long)s * HF_PAD + h * F_OUT;
    float* op = agg + (long)d * HF_PAD + h * F_OUT;
    #pragma unroll
    for (int f = 0; f < F_OUT; ++f) atomicAdd(&op[f], alpha * xs[f]);
}

// =====================================================================
// Kernel 6: elu(agg + bias) -> atomic mean-pool accumulation per graph
// =====================================================================
__global__ void gat_pool(const float* __restrict__ agg,
                         const float* __restrict__ bias,
                         const int* __restrict__ batch,
                         float* __restrict__ pool, float* __restrict__ cnt) {
    long tid = blockIdx.x * (long)blockDim.x + threadIdx.x;
    if (tid >= (long)N_NODES * HF) return;
    int n = (int)(tid / HF);
    int c = (int)(tid - (long)n * HF);
    float v = agg[(long)n * HF_PAD + c] + bias[c];
    v = v > 0.0f ? v : (expf(v) - 1.0f);            // ELU
    int g = batch[n];
    atomicAdd(&pool[(long)g * HF_PAD + c], v);
    if (c == 0) atomicAdd(&cnt[g], 1.0f);
}

// =====================================================================
// Kernel 7: h = pool / max(cnt,1); logits = h @ lin_w + lin_b; sigmoid
// out layout: [NUM_GRAPHS*HF floats of h][NUM_GRAPHS floats of sigmoid]
// =====================================================================
__global__ void gat_finalize(const float* __restrict__ pool,
                             const float* __restrict__ cnt,
                             const float* __restrict__ lin_w,
                             const float* __restrict__ lin_b,
                             float* __restrict__ out) {
    int g = blockIdx.x * blockDim.x + threadIdx.x;
    if (g >= NUM_GRAPHS) return;
    float c = cnt[g]; c = c > 1.0f ? c : 1.0f;
    float inv = 1.0f / c;
    float acc = 0.0f;
    #pragma unroll 10
    for (int j = 0; j < HF; ++j) {
        float hv = pool[(long)g * HF_PAD + j] * inv;
        out[(long)g * HF + j] = hv;
        acc += hv * lin_w[j];
    }
    float logit = acc + lin_b[0];
    out[(long)NUM_GRAPHS * HF + g] = 1.0f / (1.0f + expf(-logit));
}

// =====================================================================
extern "C" void kernel_launch(void* const* d_in, const int* in_sizes, int n_in,
                              void* d_out, int out_size, void* d_ws, size_t ws_size,
                              hipStream_t stream) {
    const float* x       = (const float*)d_in[0];
    const int*   ei      = (const int*)  d_in[1];
    const int*   batch   = (const int*)  d_in[2];
    const float* W       = (const float*)d_in[3];
    const float* att_src = (const float*)d_in[4];
    const float* att_dst = (const float*)d_in[5];
    const float* bias    = (const float*)d_in[6];
    const float* lin_w   = (const float*)d_in[7];
    const float* lin_b   = (const float*)d_in[8];
    float* out = (float*)d_out;

    // workspace carve-up (floats)
    float*    ws    = (float*)d_ws;
    float*    xw    = ws;                                   // N*64
    float*    a_s   = xw   + (long)N_NODES * HF_PAD;        // N*8
    float*    a_d   = a_s  + (long)N_NODES * H_PAD;         // N*8
    unsigned* m_u   = (unsigned*)(a_d + (long)N_NODES * H_PAD); // N*8 (ordered uint)
    float*    denom = (float*)m_u + (long)N_NODES * H_PAD;  // N*8
    float*    agg   = denom + (long)N_NODES * H_PAD;        // N*64
    float*    pool  = agg   + (long)N_NODES * HF_PAD;       // G*64
    float*    cnt   = pool  + (long)NUM_GRAPHS * HF_PAD;    // G
    long zero_count = (long)(cnt + NUM_GRAPHS - (float*)m_u);

    // 1) fp32 WMMA GEMM: xw = x @ Wcat  (6250 m-tiles x 4 n-tiles)
    gat_gemm_xw<<<N_NODES / 16, 128, 0, stream>>>(x, W, xw);

    // init atomic targets (m_u, denom, agg, pool, cnt)
    gat_zero<<<(int)((zero_count + 255) / 256), 256, 0, stream>>>((float*)m_u, zero_count);

    // 2) attention coefficients
    gat_attn<<<(N_NODES + 255) / 256, 256, 0, stream>>>(xw, att_src, att_dst, a_s, a_d);

    // 3-5) edge softmax + weighted scatter-aggregate
    gat_edge_max  <<<(E_TOT + 255) / 256, 256, 0, stream>>>(ei, a_s, a_d, m_u);
    gat_edge_denom<<<(E_TOT + 255) / 256, 256, 0, stream>>>(ei, a_s, a_d, m_u, denom);
    gat_edge_agg  <<<(int)(((long)E_TOT * HEADS + 255) / 256), 256, 0, stream>>>(
        ei, a_s, a_d, m_u, denom, xw, agg);

    // 6) ELU + mean-pool accumulation
    gat_pool<<<(int)(((long)N_NODES * HF + 255) / 256), 256, 0, stream>>>(
        agg, bias, batch, pool, cnt);

    // 7) finalize: mean, linear, sigmoid
    gat_finalize<<<(NUM_GRAPHS + 63) / 64, 64, 0, stream>>>(pool, cnt, lin_w, lin_b, out);
}